// GATv2Layer_41120016891919
// MI455X (gfx1250) — compile-verified
//
#include <hip/hip_runtime.h>
#include <hip/hip_bf16.h>

#define N_FIN   128
#define N_HF    64      // H * F_OUT
#define NEG_SLOPE 0.2f

typedef float v2f __attribute__((ext_vector_type(2)));
typedef float v8f __attribute__((ext_vector_type(8)));

// ---------------------------------------------------------------------------
// Kernel 1: h_l = X @ Wl, h_r = X @ Wr via V_WMMA_F32_16X16X4_F32.
// Block = 256 threads = 8 waves. Wave w handles column-tile w of one 16-row
// tile: w in [0,4) -> Wl/Hl col tile w ; w in [4,8) -> Wr/Hr col tile w-4.
// Each wave: 16x16 output tile, K=128 swept in steps of 4 (32 WMMA ops).
// ---------------------------------------------------------------------------
__global__ void __launch_bounds__(256)
gat_gemm_wmma_f32(const float* __restrict__ X,    // [N,128]
                  const float* __restrict__ Wl,   // [128,64]
                  const float* __restrict__ Wr,   // [128,64]
                  float* __restrict__ Hl,         // [N,64]
                  float* __restrict__ Hr,         // [N,64]
                  int N)
{
    const int lane = threadIdx.x & 31;
    const int ct   = threadIdx.x >> 5;      // 0..7 column tile (incl. l/r select)
    const int mt   = blockIdx.x;            // row tile (16 rows)

    const float* W   = (ct < 4) ? Wl : Wr;
    float*       Hout= (ct < 4) ? Hl : Hr;
    const int    nt  = ct & 3;

    const int m15 = lane & 15;              // row (A) / col (B,C,D) within tile
    const int kh  = (lane >> 4) << 1;       // 0 or 2: K sub-pair per lane half
    const int row = mt * 16 + m15;
    const int col = nt * 16 + m15;
    if (row >= N) return;

    const float* __restrict__ xrow = X + (size_t)row * N_FIN;

    v8f acc = {};
#pragma unroll
    for (int k0 = 0; k0 < N_FIN; k0 += 4) {
        v2f a, b;
        // A 16x4 layout: lanes 0-15 hold K=0 (v0), K=1 (v1); lanes 16-31 K=2,K=3
        a.x = xrow[k0 + kh];
        a.y = xrow[k0 + kh + 1];
        // B 4x16 layout (row striped across lanes): symmetric K split
        b.x = W[(k0 + kh)     * N_HF + col];
        b.y = W[(k0 + kh + 1) * N_HF + col];
        acc = __builtin_amdgcn_wmma_f32_16x16x4_f32(
                  false, a, false, b, (short)0, acc, false, false);
    }

    // C/D layout: VGPR r -> M=r (lanes 0-15) or M=r+8 (lanes 16-31); N = lane&15
    const int rbase = mt * 16 + ((lane >> 4) << 3);
#pragma unroll
    for (int r = 0; r < 8; ++r)
        Hout[(size_t)(rbase + r) * N_HF + col] = acc[r];
}

// ---------------------------------------------------------------------------
// Kernel 2: init out = bias (broadcast), m = -inf, denom = 0
// ---------------------------------------------------------------------------
__global__ void gat_init(float* __restrict__ out, float* __restrict__ mmax,
                         float* __restrict__ denom, const float* __restrict__ bias,
                         int N)
{
    int i = blockIdx.x * blockDim.x + threadIdx.x;
    if (i < N * N_HF) out[i] = bias[i & (N_HF - 1)];
    if (i < N * 4) {
        mmax[i]  = -__builtin_inff();
        denom[i] = 0.0f;
    }
}

// ---------------------------------------------------------------------------
// Kernel 3: per-edge attention scores + segment max. One wave per edge;
// lane handles features {lane, lane+32} of the 64-wide edge tile.
// Heads: elem c -> head c>>4. Butterfly-reduce within 16-lane groups.
// ---------------------------------------------------------------------------
__global__ void __launch_bounds__(256)
gat_edge_score(const float* __restrict__ Hl, const float* __restrict__ Hr,
               const int* __restrict__ src, const int* __restrict__ tgt,
               const float* __restrict__ att,
               float* __restrict__ scores,      // [E,4]
               float* __restrict__ mmax,        // [N,4]
               int E)
{
    const int e    = (blockIdx.x * blockDim.x + threadIdx.x) >> 5;
    const int lane = threadIdx.x & 31;
    if (e >= E) return;
    const int s = src[e], t = tgt[e];

    const float a0 = att[lane], a1 = att[lane + 32];
    const float* __restrict__ xi = Hl + (size_t)t * N_HF;
    const float* __restrict__ xj = Hr + (size_t)s * N_HF;

    float z0 = xi[lane]      + xj[lane];
    float z1 = xi[lane + 32] + xj[lane + 32];
    z0 = (z0 >= 0.0f) ? z0 : NEG_SLOPE * z0;
    z1 = (z1 >= 0.0f) ? z1 : NEG_SLOPE * z1;
    float p0 = z0 * a0;   // head (lane>>4):   0 (lanes 0-15) / 1 (lanes 16-31)
    float p1 = z1 * a1;   // head (lane>>4)+2: 2 / 3

#pragma unroll
    for (int off = 8; off >= 1; off >>= 1) {
        p0 += __shfl_xor(p0, off, 32);
        p1 += __shfl_xor(p1, off, 32);
    }

    if ((lane & 15) == 0) {
        const int h0 = lane >> 4;            // 0 or 1
        scores[(size_t)e * 4 + h0]     = p0;
        scores[(size_t)e * 4 + h0 + 2] = p1;
        atomicMax(&mmax[t * 4 + h0],     p0);
        atomicMax(&mmax[t * 4 + h0 + 2], p1);
    }
}

// ---------------------------------------------------------------------------
// Kernel 4: ex = exp(score - m[tgt]); scores <- ex; denom[tgt] += ex
// Flat over E*4 elements.
// ---------------------------------------------------------------------------
__global__ void gat_edge_exp(const int* __restrict__ tgt,
                             const float* __restrict__ mmax,
                             float* __restrict__ scores,
                             float* __restrict__ denom, int E)
{
    int i = blockIdx.x * blockDim.x + threadIdx.x;
    if (i >= E * 4) return;
    const int e = i >> 2, h = i & 3;
    const int t = tgt[e];
    const float ex = __expf(scores[i] - mmax[t * 4 + h]);
    scores[i] = ex;
    atomicAdd(&denom[t * 4 + h], ex);
}

// ---------------------------------------------------------------------------
// Kernel 5: alpha = ex/denom[tgt]; out[tgt] += h_r[src] * alpha. One wave/edge.
// ---------------------------------------------------------------------------
__global__ void __launch_bounds__(256)
gat_edge_agg(const float* __restrict__ Hr,
             const int* __restrict__ src, const int* __restrict__ tgt,
             const float* __restrict__ scores,   // holds ex
             const float* __restrict__ denom,
             float* __restrict__ out, int E)
{
    const int e    = (blockIdx.x * blockDim.x + threadIdx.x) >> 5;
    const int lane = threadIdx.x & 31;
    if (e >= E) return;
    const int s = src[e], t = tgt[e];

    const int h0 = lane >> 4;        // head of elem lane
    const float alpha0 = scores[(size_t)e * 4 + h0]     / denom[t * 4 + h0];
    const float alpha1 = scores[(size_t)e * 4 + h0 + 2] / denom[t * 4 + h0 + 2];

    const float* __restrict__ xj = Hr + (size_t)s * N_HF;
    atomicAdd(&out[(size_t)t * N_HF + lane],      xj[lane]      * alpha0);
    atomicAdd(&out[(size_t)t * N_HF + lane + 32], xj[lane + 32] * alpha1);
}

// ---------------------------------------------------------------------------
extern "C" void kernel_launch(void* const* d_in, const int* in_sizes, int n_in,
                              void* d_out, int out_size, void* d_ws, size_t ws_size,
                              hipStream_t stream)
{
    const float* features = (const float*)d_in[0];
    const int*   edge_idx = (const int*)  d_in[1];
    const float* weight_l = (const float*)d_in[2];
    const float* weight_r = (const float*)d_in[3];
    const float* att      = (const float*)d_in[4];
    const float* bias     = (const float*)d_in[5];
    float*       out      = (float*)d_out;

    const int N = in_sizes[0] / N_FIN;    // 100000
    const int E = in_sizes[1] / 2;        // 1600000
    const int* src = edge_idx;
    const int* tgt = edge_idx + E;

    // Workspace layout (floats): Hl | Hr | scores | m | denom  (~80 MB)
    float* ws     = (float*)d_ws;
    float* Hl     = ws;
    float* Hr     = Hl     + (size_t)N * N_HF;
    float* scores = Hr     + (size_t)N * N_HF;
    float* mmax   = scores + (size_t)E * 4;
    float* denom  = mmax   + (size_t)N * 4;

    // 1) dense GEMMs via f32 WMMA: one block of 8 waves per 16-row tile
    {
        dim3 grid((N + 15) / 16);
        gat_gemm_wmma_f32<<<grid, 256, 0, stream>>>(features, weight_l, weight_r,
                                                    Hl, Hr, N);
    }
    // 2) init out=bias, m=-inf, denom=0
    {
        int total = N * N_HF;
        gat_init<<<(total + 255) / 256, 256, 0, stream>>>(out, mmax, denom, bias, N);
    }
    // 3) edge scores + segment max (1 wave / edge)
    {
        long long waves = E;
        gat_edge_score<<<(unsigned)((waves + 7) / 8), 256, 0, stream>>>(
            Hl, Hr, src, tgt, att, scores, mmax, E);
    }
    // 4) exp + segment sum
    {
        long long total = (long long)E * 4;
        gat_edge_exp<<<(unsigned)((total + 255) / 256), 256, 0, stream>>>(
            tgt, mmax, scores, denom, E);
    }
    // 5) normalize + weighted scatter-add (1 wave / edge)
    {
        long long waves = E;
        gat_edge_agg<<<(unsigned)((waves + 7) / 8), 256, 0, stream>>>(
            Hr, src, tgt, scores, denom, out, E);
    }
}